// GATEncoder_71305047048703
// MI455X (gfx1250) — compile-verified
//
#include <hip/hip_runtime.h>
#include <hip/hip_bf16.h>

// ---------------- problem constants (from the reference) ----------------
#define NN      50000
#define EE      800000
#define IN_DIM  64
#define HIDDEN  128
#define HEADS   4
#define HEAD_DIM 32
#define EDGE_DIM 6
#define NEG_SLOPE 0.2f
#define BN_EPS  1e-5f

typedef __attribute__((ext_vector_type(2))) float v2f;
typedef __attribute__((ext_vector_type(8))) float v8f;

// ---------------------------------------------------------------------------
// float atomic max via monotonic int mapping (no NaNs in this workload)
// ---------------------------------------------------------------------------
__device__ __forceinline__ void atomicMaxF32(float* addr, float v) {
    int iv = __float_as_int(v);
    if (iv >= 0) {
        atomicMax((int*)addr, iv);
    } else {
        atomicMin((unsigned int*)addr, (unsigned int)iv);
    }
}

// ---------------------------------------------------------------------------
// GEMM: out[M,128] = act(A[M,K] @ W[K,128] + bias), K % 4 == 0, M % 16 == 0.
// One 16x16 output tile per wave via V_WMMA_F32_16X16X4_F32.
// Block = 256 threads = 8 waves -> 16 rows x 128 cols per block.
// Fragment layouts per CDNA5 ISA 7.12.2:
//   A (16x4):  lanes 0-15 row=lane, v0=K0,v1=K1 ; lanes 16-31 row=lane-16, v0=K2,v1=K3
//   B (4x16):  lanes 0-15 col=lane, v0=K0,v1=K1 ; lanes 16-31 col=lane-16, v0=K2,v1=K3
//   C (16x16): vgpr i: lanes 0-15 -> M=i, lanes 16-31 -> M=8+i; N = lane&15
// ---------------------------------------------------------------------------
__global__ __launch_bounds__(256)
void gemm_wmma_kernel(const float* __restrict__ A, const float* __restrict__ W,
                      const float* __restrict__ bias, float* __restrict__ out,
                      int K, int act) {
    const int lane = threadIdx.x & 31;
    const int wv   = threadIdx.x >> 5;
    const int half = lane >> 4;      // 0: K pair {0,1}, 1: K pair {2,3}
    const int l15  = lane & 15;
    const int row0 = blockIdx.x * 16;
    const int col0 = wv * 16;

    v8f c = {};
    const float* arow = A + (size_t)(row0 + l15) * K + half * 2;
    for (int k = 0; k < K; k += 4) {
        v2f a, b;
        a.x = arow[k];
        a.y = arow[k + 1];
        const float* bp = W + (size_t)(k + half * 2) * HIDDEN + col0 + l15;
        b.x = bp[0];
        b.y = bp[HIDDEN];
        c = __builtin_amdgcn_wmma_f32_16x16x4_f32(
                /*neg_a=*/false, a, /*neg_b=*/false, b,
                /*c_mod=*/(short)0, c, /*reuse_a=*/false, /*reuse_b=*/false);
    }

    const int cc = col0 + l15;
    const float bv = bias[cc];
#pragma unroll
    for (int i = 0; i < 8; ++i) {
        float v = c[i] + bv;
        if (act) v = fmaxf(v, 0.0f);
        out[(size_t)(row0 + half * 8 + i) * HIDDEN + cc] = v;
    }
}

// ---------------------------------------------------------------------------
// Init per-layer accumulators: acc=0, nmax=-inf, denom=0
// grid*block covers NN*HIDDEN
// ---------------------------------------------------------------------------
__global__ __launch_bounds__(256)
void init_kernel(float* __restrict__ acc, float* __restrict__ nmax,
                 float* __restrict__ denom) {
    int t = blockIdx.x * 256 + threadIdx.x;
    if (t < NN * HIDDEN) acc[t] = 0.0f;
    if (t < NN * HEADS) {
        nmax[t]  = __int_as_float(0xFF800000); // -inf
        denom[t] = 0.0f;
    }
}

// ---------------------------------------------------------------------------
// Edge logits: one wave per edge. 4 channels/lane, heads = lane/8.
//   m = leakyrelu(xl[src] + xr[dst] + (ea@We + be)); logit[h] = sum_c m*att
// Writes logits[E,4] and atomic-max into nmax[N,4].
// ---------------------------------------------------------------------------
__global__ __launch_bounds__(256)
void edge_logits_kernel(const float* __restrict__ xl, const float* __restrict__ xr,
                        const int* __restrict__ ei, const float* __restrict__ ea,
                        const float* __restrict__ We, const float* __restrict__ be,
                        const float* __restrict__ att,
                        float* __restrict__ logits, float* __restrict__ nmax) {
    __shared__ float sWe[EDGE_DIM * HIDDEN];
    __shared__ float sbe[HIDDEN];
    __shared__ float satt[HIDDEN];
    for (int i = threadIdx.x; i < EDGE_DIM * HIDDEN; i += 256) sWe[i] = We[i];
    for (int i = threadIdx.x; i < HIDDEN; i += 256) { sbe[i] = be[i]; satt[i] = att[i]; }
    __syncthreads();

    const int wv   = threadIdx.x >> 5;
    const int lane = threadIdx.x & 31;
    const int e    = blockIdx.x * 8 + wv;           // EE % 8 == 0
    const int src  = __builtin_amdgcn_readfirstlane(ei[e]);
    const int dst  = __builtin_amdgcn_readfirstlane(ei[EE + e]);

    const int c0 = lane * 4;
    const float4 a = *(const float4*)(xl + (size_t)src * HIDDEN + c0);
    const float4 b = *(const float4*)(xr + (size_t)dst * HIDDEN + c0);

    float ew[EDGE_DIM];
#pragma unroll
    for (int d = 0; d < EDGE_DIM; ++d) ew[d] = ea[(size_t)e * EDGE_DIM + d];

    const float av[4] = {a.x, a.y, a.z, a.w};
    const float bv[4] = {b.x, b.y, b.z, b.w};
    float s = 0.0f;
#pragma unroll
    for (int k = 0; k < 4; ++k) {
        const int c = c0 + k;
        float m = av[k] + bv[k] + sbe[c];
#pragma unroll
        for (int d = 0; d < EDGE_DIM; ++d) m += ew[d] * sWe[d * HIDDEN + c];
        m = (m > 0.0f) ? m : NEG_SLOPE * m;         // leaky relu
        s += m * satt[c];
    }
    // reduce across the 8 lanes that make up one head
    s += __shfl_xor(s, 1, 8);
    s += __shfl_xor(s, 2, 8);
    s += __shfl_xor(s, 4, 8);

    if ((lane & 7) == 0) {
        const int h = lane >> 3;
        logits[e * HEADS + h] = s;
        atomicMaxF32(&nmax[dst * HEADS + h], s);
    }
}

// ---------------------------------------------------------------------------
// p = exp(logit - nmax[dst]); store p in place; denom[dst] += p
// one thread per (edge, head); grid*block == EE*HEADS
// ---------------------------------------------------------------------------
__global__ __launch_bounds__(256)
void edge_exp_kernel(float* __restrict__ plog, const float* __restrict__ nmax,
                     float* __restrict__ denom, const int* __restrict__ ei) {
    const int t = blockIdx.x * 256 + threadIdx.x;   // < EE*HEADS
    const int e = t >> 2;
    const int h = t & 3;
    const int dst = ei[EE + e];
    const float p = __expf(plog[t] - nmax[dst * HEADS + h]);
    plog[t] = p;
    atomicAdd(&denom[dst * HEADS + h], p);
}

// ---------------------------------------------------------------------------
// acc[dst] += xl[src] * alpha  with alpha = p / denom[dst]; wave per edge
// ---------------------------------------------------------------------------
__global__ __launch_bounds__(256)
void edge_scatter_kernel(const float* __restrict__ xl, const float* __restrict__ plog,
                         const float* __restrict__ denom, const int* __restrict__ ei,
                         float* __restrict__ acc) {
    const int wv   = threadIdx.x >> 5;
    const int lane = threadIdx.x & 31;
    const int e    = blockIdx.x * 8 + wv;
    const int src  = __builtin_amdgcn_readfirstlane(ei[e]);
    const int dst  = __builtin_amdgcn_readfirstlane(ei[EE + e]);
    const int h    = lane >> 3;
    const float alpha = plog[e * HEADS + h] / denom[dst * HEADS + h];
    const int c0 = lane * 4;
    const float4 v = *(const float4*)(xl + (size_t)src * HIDDEN + c0);
    float* base = acc + (size_t)dst * HIDDEN + c0;
    atomicAdd(base + 0, v.x * alpha);
    atomicAdd(base + 1, v.y * alpha);
    atomicAdd(base + 2, v.z * alpha);
    atomicAdd(base + 3, v.w * alpha);
}

// ---------------------------------------------------------------------------
// BatchNorm stats: one block per channel; fold (gamma, beta, mu, var) into
// scale/shift:  y = v*scale + shift,  v = acc + conv_bias
// ---------------------------------------------------------------------------
__global__ __launch_bounds__(256)
void bn_stats_kernel(const float* __restrict__ acc, const float* __restrict__ cbias,
                     const float* __restrict__ gamma, const float* __restrict__ beta,
                     float* __restrict__ scale, float* __restrict__ shift) {
    const int c = blockIdx.x;                       // 0..127
    const float b = cbias[c];
    float s = 0.0f, sq = 0.0f;
    for (int n = threadIdx.x; n < NN; n += 256) {
        const float v = acc[(size_t)n * HIDDEN + c] + b;
        s += v;
        sq += v * v;
    }
    __shared__ float ss[256], ssq[256];
    ss[threadIdx.x] = s;
    ssq[threadIdx.x] = sq;
    __syncthreads();
    for (int st = 128; st > 0; st >>= 1) {
        if (threadIdx.x < st) {
            ss[threadIdx.x]  += ss[threadIdx.x + st];
            ssq[threadIdx.x] += ssq[threadIdx.x + st];
        }
        __syncthreads();
    }
    if (threadIdx.x == 0) {
        const float mu  = ss[0] / (float)NN;
        const float var = ssq[0] / (float)NN - mu * mu;
        const float sc  = gamma[c] * rsqrtf(var + BN_EPS);
        scale[c] = sc;
        shift[c] = beta[c] - mu * sc;
    }
}

// ---------------------------------------------------------------------------
// out = relu((acc + conv_bias)*scale + shift)
// ---------------------------------------------------------------------------
__global__ __launch_bounds__(256)
void bn_apply_kernel(const float* __restrict__ acc, const float* __restrict__ cbias,
                     const float* __restrict__ scale, const float* __restrict__ shift,
                     float* __restrict__ out) {
    const int t = blockIdx.x * 256 + threadIdx.x;   // < NN*HIDDEN
    const int c = t & (HIDDEN - 1);
    float v = acc[t] + cbias[c];
    v = v * scale[c] + shift[c];
    out[t] = fmaxf(v, 0.0f);
}

// ---------------------------------------------------------------------------
// Launcher
// ---------------------------------------------------------------------------
extern "C" void kernel_launch(void* const* d_in, const int* in_sizes, int n_in,
                              void* d_out, int out_size, void* d_ws, size_t ws_size,
                              hipStream_t stream) {
    const float* x     = (const float*)d_in[0];
    const int*   ei    = (const int*)  d_in[1];
    const float* ea    = (const float*)d_in[2];
    const float* Wproj = (const float*)d_in[3];
    const float* bproj = (const float*)d_in[4];
    const float* Wl    = (const float*)d_in[5];
    const float* bl    = (const float*)d_in[6];
    const float* Wr    = (const float*)d_in[7];
    const float* br    = (const float*)d_in[8];
    const float* We    = (const float*)d_in[9];
    const float* be    = (const float*)d_in[10];
    const float* att   = (const float*)d_in[11];
    const float* cb    = (const float*)d_in[12];
    const float* gamma = (const float*)d_in[13];
    const float* beta  = (const float*)d_in[14];
    float* out = (float*)d_out;

    // workspace layout (floats)
    float* ws    = (float*)d_ws;
    float* h     = ws;                                   // NN*HIDDEN
    float* xl    = h     + (size_t)NN * HIDDEN;          // NN*HIDDEN
    float* xr    = xl    + (size_t)NN * HIDDEN;          // NN*HIDDEN
    float* acc   = xr    + (size_t)NN * HIDDEN;          // NN*HIDDEN
    float* plog  = acc   + (size_t)NN * HIDDEN;          // EE*HEADS
    float* nmax  = plog  + (size_t)EE * HEADS;           // NN*HEADS
    float* denom = nmax  + (size_t)NN * HEADS;           // NN*HEADS
    float* bnsc  = denom + (size_t)NN * HEADS;           // HIDDEN
    float* bnsh  = bnsc  + HIDDEN;                       // HIDDEN

    const int gemm_grid = NN / 16;                       // 3125
    const int elem_grid = (NN * HIDDEN) / 256;           // 25000
    const int edge_grid = EE / 8;                        // 100000
    const int eh_grid   = (EE * HEADS) / 256;            // 12500

    // input projection + relu: h = relu(x @ Wproj + bproj)
    gemm_wmma_kernel<<<gemm_grid, 256, 0, stream>>>(x, Wproj, bproj, h, IN_DIM, 1);

    for (int l = 0; l < 2; ++l) {
        const float* Wl_l = Wl + (size_t)l * HIDDEN * HIDDEN;
        const float* Wr_l = Wr + (size_t)l * HIDDEN * HIDDEN;
        const float* We_l = We + (size_t)l * EDGE_DIM * HIDDEN;

        gemm_wmma_kernel<<<gemm_grid, 256, 0, stream>>>(h, Wl_l, bl + l * HIDDEN, xl, HIDDEN, 0);
        gemm_wmma_kernel<<<gemm_grid, 256, 0, stream>>>(h, Wr_l, br + l * HIDDEN, xr, HIDDEN, 0);

        init_kernel<<<elem_grid, 256, 0, stream>>>(acc, nmax, denom);

        edge_logits_kernel<<<edge_grid, 256, 0, stream>>>(
            xl, xr, ei, ea, We_l, be + l * HIDDEN, att + l * HIDDEN, plog, nmax);
        edge_exp_kernel<<<eh_grid, 256, 0, stream>>>(plog, nmax, denom, ei);
        edge_scatter_kernel<<<edge_grid, 256, 0, stream>>>(xl, plog, denom, ei, acc);

        bn_stats_kernel<<<HIDDEN, 256, 0, stream>>>(
            acc, cb + l * HIDDEN, gamma + l * HIDDEN, beta + l * HIDDEN, bnsc, bnsh);
        bn_apply_kernel<<<elem_grid, 256, 0, stream>>>(
            acc, cb + l * HIDDEN, bnsc, bnsh, (l == 0) ? h : out);
    }
}